// FSRS_18597208391855
// MI455X (gfx1250) — compile-verified
//
#include <hip/hip_runtime.h>

#define FSRS_S_MIN 0.01f
#define FSRS_S_MAX 36500.0f
#define FSRS_SEQ_LEN 128
#define FSRS_BATCH 65536

// Native clang vector so nontemporal builtins accept it (HIP float2 is a class).
typedef __attribute__((ext_vector_type(2))) float f32x2;

// fast pow for strictly-positive base: exp(e*log(b)) -> v_log_f32 + v_exp_f32
__device__ __forceinline__ float fsrs_powf(float base, float e) {
    return __expf(e * __logf(base));
}

__global__ __launch_bounds__(256) void fsrs_scan_kernel(
    const float* __restrict__ inp,   // (SEQ_LEN, BATCH, 2)
    const float* __restrict__ w,     // (17,)
    float*       __restrict__ out)   // (SEQ_LEN, BATCH, 2) ++ (BATCH, 2)
{
    const int b = blockIdx.x * blockDim.x + threadIdx.x;   // 0..BATCH-1

    // Uniform weight loads (scalar path) + hoisted derived constants.
    const float w0  = w[0],  w1  = w[1],  w2  = w[2],  w3  = w[3];
    const float w4  = w[4],  w5  = w[5],  w6  = w[6],  w7  = w[7];
    const float w9  = w[9],  w10 = w[10], w11 = w[11], w12 = w[12];
    const float w13 = w[13], w14 = w[14], w15 = w[15], w16 = w[16];
    const float ew8      = __expf(w[8]);
    const float mean_rev = w7 * w4;          // w7*w4
    const float omw7     = 1.0f - w7;        // (1 - w7)

    const f32x2* __restrict__ in2  = reinterpret_cast<const f32x2*>(inp);
    f32x2*       __restrict__ out2 = reinterpret_cast<f32x2*>(out);
    f32x2*       __restrict__ fin2 =
        reinterpret_cast<f32x2*>(out + (size_t)FSRS_SEQ_LEN * FSRS_BATCH * 2);

    // ---------- first step (t = 0) ----------
    float s, d;
    {
        f32x2 x = __builtin_nontemporal_load(&in2[b]);
        const float rating = x.y;
        int ri = (int)rating - 1;
        ri = ri < 0 ? 0 : (ri > 3 ? 3 : ri);
        float sv = w0;
        sv = (ri == 1) ? w1 : sv;
        sv = (ri == 2) ? w2 : sv;
        sv = (ri == 3) ? w3 : sv;
        const bool valid = (rating >= 1.0f) && (rating <= 4.0f);
        s = valid ? sv : 1.0f;
        d = __fmaf_rn(-w5, rating - 3.0f, w4);
        d = fminf(fmaxf(d, 1.0f), 10.0f);
        s = fminf(fmaxf(s, FSRS_S_MIN), FSRS_S_MAX);
        f32x2 st; st.x = s; st.y = d;
        __builtin_nontemporal_store(st, &out2[b]);
    }

    // ---------- scan (t = 1 .. SEQ_LEN-1) ----------
#pragma unroll 2
    for (int t = 1; t < FSRS_SEQ_LEN; ++t) {
        f32x2 x = __builtin_nontemporal_load(&in2[(size_t)t * FSRS_BATCH + b]);

        // gfx1250 prefetch path: pull a future timestep's line toward GL2/L0.
        if (t + 8 < FSRS_SEQ_LEN) {
            __builtin_prefetch(&in2[(size_t)(t + 8) * FSRS_BATCH + b], 0, 1);
        }

        const float tv     = x.x;
        const float rating = x.y;

        // r = 1/(1 + t/(9s))  =>  (1-r) = t / (9s + t)
        const float denom = __fmaf_rn(9.0f, s, tv);
        const float omr   = tv * __frcp_rn(denom);         // (1 - r)

        // branchless hard-penalty / easy-bonus
        float hpeb = 1.0f;
        hpeb = (rating == 2.0f) ? w15 : hpeb;
        hpeb = (rating == 4.0f) ? (hpeb * w16) : hpeb;     // rating can't be both

        // success branch
        const float pow_s = fsrs_powf(s, -w9);             // s^-w9
        const float grow  = __expf(omr * w10) - 1.0f;      // e^{(1-r)w10} - 1
        const float mult  = ew8 * (11.0f - d) * pow_s * grow * hpeb;
        const float s_succ = __fmaf_rn(s, mult, s);        // s*(1+mult)

        // failure branch
        const float pow_d  = fsrs_powf(d, -w12);           // d^-w12
        const float pow_sp = fsrs_powf(s + 1.0f, w13) - 1.0f;
        const float s_fail = fminf(w11 * pow_d * pow_sp * __expf(omr * w14), s);

        float new_s = (rating > 1.0f) ? s_succ : s_fail;
        new_s = fminf(fmaxf(new_s, FSRS_S_MIN), FSRS_S_MAX);

        // difficulty update with mean reversion
        float nd = __fmaf_rn(-w6, rating - 3.0f, d);       // d - w6*(rating-3)
        nd = __fmaf_rn(omw7, nd, mean_rev);                // w7*w4 + (1-w7)*nd
        nd = fminf(fmaxf(nd, 1.0f), 10.0f);

        s = new_s;
        d = nd;

        f32x2 st; st.x = s; st.y = d;
        __builtin_nontemporal_store(st, &out2[(size_t)t * FSRS_BATCH + b]);
    }

    // ---------- final state ----------
    {
        f32x2 st; st.x = s; st.y = d;
        __builtin_nontemporal_store(st, &fin2[b]);
    }
}

extern "C" void kernel_launch(void* const* d_in, const int* in_sizes, int n_in,
                              void* d_out, int out_size, void* d_ws, size_t ws_size,
                              hipStream_t stream) {
    (void)in_sizes; (void)n_in; (void)out_size; (void)d_ws; (void)ws_size;
    const float* inp = (const float*)d_in[0];   // (128, 65536, 2) float32
    const float* w   = (const float*)d_in[1];   // (17,) float32
    float* out       = (float*)d_out;           // outputs ++ final_state

    const int block = 256;                      // 8 wave32 waves per block
    const int grid  = FSRS_BATCH / block;       // 256 blocks
    fsrs_scan_kernel<<<grid, block, 0, stream>>>(inp, w, out);
}